// AffineContour_4183298146930
// MI455X (gfx1250) — compile-verified
//
#include <hip/hip_runtime.h>

// ---------------------------------------------------------------------------
// AffineContour on MI455X (gfx1250, wave32)
//
// Structure:
//   K1: fp32 GEMV (x_even @ {u,v}_W0) via V_WMMA_F32_16X16X4_F32, partials->ws
//       (dual accumulators -> two independent WMMA chains per wave)
//   K2: reduce+bias+relu -> layer2 GEMV -> relu -> scalar heads u_s,u_t -> ws
//   K3: complex scatter: out[i] = x[i] + 1j*(odd ? u_s*x[i]+u_t : 0)
// ---------------------------------------------------------------------------

#define LAT_V   131072
#define LAT_VH  65536
#define FEAT    512
#define NTILES  (FEAT / 16)     // 32 column tiles of 16 features
#define NKB     64              // K blocks per matrix (4096 waves total)
#define KCHUNK  (LAT_VH / NKB)  // 1024 rows per K block

typedef __attribute__((ext_vector_type(2))) float v2f;
typedef __attribute__((ext_vector_type(8))) float v8f;

// ---------------------------------------------------------------------------
// Kernel 1: layer-1 GEMV partials with fp32 WMMA.
// One wave (32 threads) per block. Each block owns (matrix, n-tile, k-block).
//
// WMMA D = A x B + C, 16x16x4 fp32:
//   A (16x4): lanes 0-15 hold {K0,K1}, lanes 16-31 hold {K2,K3}; M = lane%16.
//     We broadcast the same 4 x-values into every row M => each lane's pair is
//     {x[k+2*hi], x[k+2*hi+1]} with hi = lane>>4.
//   B (4x16): striped like C/D: VGPR0 = row K0 (lanes 0-15) / K2 (lanes
//     16-31), VGPR1 = K1 / K3, N = lane%16.
//   D (16x16): all rows equal => VGPR0 on lanes 0-15 gives features n=0..15.
// ---------------------------------------------------------------------------
__global__ __launch_bounds__(32)
void k1_layer1_wmma(const float* __restrict__ x,
                    const int*   __restrict__ even_idx,
                    const float* __restrict__ u_W0,
                    const float* __restrict__ v_W0,
                    float* __restrict__ partial /* [2][NKB][FEAT] */) {
    const int lane = threadIdx.x;      // 0..31, full wave, no divergence
    const int hi   = lane >> 4;        // 0 or 1 (half-wave)
    const int nl   = lane & 15;

    int bid = blockIdx.x;              // 0 .. 2*NTILES*NKB-1
    const int mat   = bid / (NTILES * NKB);
    bid            -= mat * (NTILES * NKB);
    const int ntile = bid / NKB;
    const int kb    = bid - ntile * NKB;

    const float* __restrict__ W = mat ? v_W0 : u_W0;
    const int k0   = kb * KCHUNK;
    const int ncol = ntile * 16 + nl;

    // Two accumulators -> two independent WMMA dependency chains.
    v8f c0 = {};
    v8f c1 = {};
    #pragma unroll 4
    for (int k = k0; k < k0 + KCHUNK; k += 8) {
        const int ka0 = k + 2 * hi;        // chain 0: rows k .. k+3
        const int ka1 = k + 4 + 2 * hi;    // chain 1: rows k+4 .. k+7

        v2f a0, b0, a1, b1;
        // A: x chunk broadcast to all 16 rows (gather through even_indices)
        a0.x = x[even_idx[ka0]];
        a0.y = x[even_idx[ka0 + 1]];
        a1.x = x[even_idx[ka1]];
        a1.y = x[even_idx[ka1 + 1]];
        // B: 4x16 tiles of W0 (row-major, stride FEAT floats)
        b0.x = W[(size_t)ka0 * FEAT + ncol];
        b0.y = W[(size_t)(ka0 + 1) * FEAT + ncol];
        b1.x = W[(size_t)ka1 * FEAT + ncol];
        b1.y = W[(size_t)(ka1 + 1) * FEAT + ncol];

        // (neg_a, A, neg_b, B, c_mod, C, reuse_a, reuse_b)
        c0 = __builtin_amdgcn_wmma_f32_16x16x4_f32(
                 false, a0, false, b0, (short)0, c0, false, false);
        c1 = __builtin_amdgcn_wmma_f32_16x16x4_f32(
                 false, a1, false, b1, (short)0, c1, false, false);
    }
    // All 16 D rows are identical; row M=0 lives in element 0 on lanes 0-15.
    if (lane < 16) {
        partial[((size_t)mat * NKB + kb) * FEAT + ncol] = c0[0] + c1[0];
    }
}

// ---------------------------------------------------------------------------
// Kernel 2: single block, 512 threads.
//   h  = relu(sum_kb partial + b0)           (both u and v)
//   h2 = relu(h @ W1 + b1)
//   u_s = h2_u . us_W + us_b ; u_t = h2_v . ut_W + ut_b   -> scalars[0..1]
// ---------------------------------------------------------------------------
__global__ __launch_bounds__(FEAT)
void k2_layer2_heads(const float* __restrict__ partial,
                     const float* __restrict__ u_b0,
                     const float* __restrict__ u_W1,
                     const float* __restrict__ u_b1,
                     const float* __restrict__ v_b0,
                     const float* __restrict__ v_W1,
                     const float* __restrict__ v_b1,
                     const float* __restrict__ us_W,
                     const float* __restrict__ us_b,
                     const float* __restrict__ ut_W,
                     const float* __restrict__ ut_b,
                     float* __restrict__ scalars /* [2] */) {
    __shared__ float hu[FEAT];
    __shared__ float hv[FEAT];
    __shared__ float red0[FEAT];
    __shared__ float red1[FEAT];

    const int n = threadIdx.x;

    // reduce K partials, add bias, relu
    float au = u_b0[n];
    float av = v_b0[n];
    #pragma unroll 8
    for (int kb = 0; kb < NKB; ++kb) {
        au += partial[((size_t)0 * NKB + kb) * FEAT + n];
        av += partial[((size_t)1 * NKB + kb) * FEAT + n];
    }
    hu[n] = fmaxf(au, 0.0f);
    hv[n] = fmaxf(av, 0.0f);
    __syncthreads();

    // layer 2 GEMV: column n of W1, coalesced across threads
    float bu = u_b1[n];
    float bv = v_b1[n];
    for (int k = 0; k < FEAT; ++k) {
        bu = fmaf(hu[k], u_W1[(size_t)k * FEAT + n], bu);
        bv = fmaf(hv[k], v_W1[(size_t)k * FEAT + n], bv);
    }
    const float u2 = fmaxf(bu, 0.0f);
    const float v2 = fmaxf(bv, 0.0f);

    // scalar heads: block tree reduction
    red0[n] = u2 * us_W[n];
    red1[n] = v2 * ut_W[n];
    __syncthreads();
    for (int s = FEAT / 2; s > 0; s >>= 1) {
        if (n < s) {
            red0[n] += red0[n + s];
            red1[n] += red1[n + s];
        }
        __syncthreads();
    }
    if (n == 0) {
        scalars[0] = red0[0] + us_b[0];   // u_s
        scalars[1] = red1[0] + ut_b[0];   // u_t
    }
}

// ---------------------------------------------------------------------------
// Kernel 3: complex affine scatter. One thread per (even,odd) site pair.
//   out[i] = x[i] + 1j*(i odd ? u_s*x[i]+u_t : 0), interleaved complex64.
// ---------------------------------------------------------------------------
__global__ __launch_bounds__(256)
void k3_scatter(const float* __restrict__ x,
                const int*   __restrict__ even_idx,
                const int*   __restrict__ odd_idx,
                const float* __restrict__ scalars,
                float2* __restrict__ out) {
    const int j = blockIdx.x * blockDim.x + threadIdx.x;
    if (j >= LAT_VH) return;
    const float us = scalars[0];
    const float ut = scalars[1];
    const int ie = even_idx[j];
    const int io = odd_idx[j];
    const float xe = x[ie];
    const float xo = x[io];
    out[ie] = make_float2(xe, 0.0f);
    out[io] = make_float2(xo, fmaf(us, xo, ut));
}

// ---------------------------------------------------------------------------
extern "C" void kernel_launch(void* const* d_in, const int* in_sizes, int n_in,
                              void* d_out, int out_size, void* d_ws, size_t ws_size,
                              hipStream_t stream) {
    // setup_inputs() dict order
    const float* x        = (const float*)d_in[0];
    const int*   even_idx = (const int*)  d_in[1];
    const int*   odd_idx  = (const int*)  d_in[2];
    const float* u_W0     = (const float*)d_in[3];
    const float* u_b0     = (const float*)d_in[4];
    const float* u_W1     = (const float*)d_in[5];
    const float* u_b1     = (const float*)d_in[6];
    const float* v_W0     = (const float*)d_in[7];
    const float* v_b0     = (const float*)d_in[8];
    const float* v_W1     = (const float*)d_in[9];
    const float* v_b1     = (const float*)d_in[10];
    const float* us_W     = (const float*)d_in[11];
    const float* us_b     = (const float*)d_in[12];
    const float* ut_W     = (const float*)d_in[13];
    const float* ut_b     = (const float*)d_in[14];

    // workspace layout: [2][NKB][FEAT] partials (256 KB), then 2 scalars
    float* partial = (float*)d_ws;
    float* scalars = partial + (size_t)2 * NKB * FEAT;

    // K1: 2 matrices x 32 n-tiles x 64 k-blocks, one wave each (4096 waves)
    k1_layer1_wmma<<<2 * NTILES * NKB, 32, 0, stream>>>(
        x, even_idx, u_W0, v_W0, partial);

    // K2: single block, 512 threads
    k2_layer2_heads<<<1, FEAT, 0, stream>>>(
        partial, u_b0, u_W1, u_b1, v_b0, v_W1, v_b1,
        us_W, us_b, ut_W, ut_b, scalars);

    // K3: one thread per site pair
    k3_scatter<<<(LAT_VH + 255) / 256, 256, 0, stream>>>(
        x, even_idx, odd_idx, scalars, (float2*)d_out);
}